// SemanticEmbedding_68899865362565
// MI455X (gfx1250) — compile-verified
//
#include <hip/hip_runtime.h>
#include <hip/hip_bf16.h>

// out[token, :] = (ids[token] == PADDING_IDX) ? (W[PADDING_IDX] + b) : float(PADDING_IDX)
// PADDING_IDX == 0 -> common path writes zeros, rare path (p ~= 1/50257) writes W[0]+b.
// Pure store-bandwidth-bound: 100.7 MB fp32 output, ~4.3 us floor at 23.3 TB/s.
// Steady state per wave32: 1 uniform id load + 6x global_store_b128 (NT) per lane.

#define EMBED_DIM   768
#define D4          (EMBED_DIM / 4)   // 192 float4 chunks per row
#define PADDING_IDX 0

typedef __attribute__((ext_vector_type(4))) float v4f;

__global__ __launch_bounds__(256) void SemanticEmbedding_68899865362565_kernel(
    const int* __restrict__ ids,
    const float* __restrict__ W,
    const float* __restrict__ b,
    float* __restrict__ out,
    int n_tokens)
{
    // wave32 tiling: one wave per 3 KB token row, 6 x float4 NT stores per lane.
    const int lane            = threadIdx.x & 31;
    const int warp            = threadIdx.x >> 5;
    const int warps_per_block = blockDim.x >> 5;

    const int token = blockIdx.x * warps_per_block + warp;
    if (token >= n_tokens) return;

    const int id = ids[token];   // uniform within the wave -> scalar branch below

    v4f* __restrict__ dst = reinterpret_cast<v4f*>(out) + (size_t)token * D4;

    if (id != PADDING_IDX) {
        // Common path: constant float(PADDING_IDX) == 0.0f vector. No loads at all.
        const v4f z = (v4f)((float)PADDING_IDX);
        #pragma unroll
        for (int k = 0; k < 6; ++k) {
            const int c = lane + 32 * k;
            __builtin_nontemporal_store(z, &dst[c]);
        }
    } else {
        // Rare path (~0.65 tokens expected per whole launch): W[0]+b from global.
        // Row 0 of W and b are 6 KB total and stay L2-resident.
        const v4f* __restrict__ W4 =
            reinterpret_cast<const v4f*>(W + (size_t)PADDING_IDX * EMBED_DIM);
        const v4f* __restrict__ b4 = reinterpret_cast<const v4f*>(b);
        #pragma unroll
        for (int k = 0; k < 6; ++k) {
            const int c = lane + 32 * k;
            __builtin_nontemporal_store(W4[c] + b4[c], &dst[c]);
        }
    }
}

extern "C" void kernel_launch(void* const* d_in, const int* in_sizes, int n_in,
                              void* d_out, int out_size, void* d_ws, size_t ws_size,
                              hipStream_t stream)
{
    (void)n_in; (void)out_size; (void)d_ws; (void)ws_size;

    const int*   ids = (const int*)  d_in[0];   // [4, 8192] int32
    const float* W   = (const float*)d_in[1];   // [50257, 768] fp32
    const float* b   = (const float*)d_in[2];   // [768] fp32
    float*       out = (float*)      d_out;     // [4, 8192, 768] fp32

    const int n_tokens        = in_sizes[0];    // 32768
    const int block           = 256;            // 8 wave32s
    const int warps_per_block = block / 32;
    const int grid            = (n_tokens + warps_per_block - 1) / warps_per_block;

    SemanticEmbedding_68899865362565_kernel<<<grid, block, 0, stream>>>(ids, W, b, out, n_tokens);
}